// NGramRepeatBlock_82652350644921
// MI455X (gfx1250) — compile-verified
//
#include <hip/hip_runtime.h>
#include <hip/hip_bf16.h>

// NGramRepeatBlock for MI455X (gfx1250, wave32).
//
// Roofline: the op is a 206 MB streaming copy of lprobs (B=512 x V=50257 f32)
// plus a ~2 MB token scan with a few scattered -inf stores. Pure HBM bandwidth
// (23.3 TB/s -> ~9us floor); no matrix math exists in this op, so no WMMA.
// gfx1250-specific paths used: 128-bit VMEM with non-temporal TH hints
// (don't thrash 192MB L2 with read-once data), global_prefetch_b8, and LDS
// staging of the token row. This revision unrolls the copy x4 with
// independent registers so each wave keeps 4 global_load_b128 in flight
// (previous codegen serialized on s_wait_loadcnt 0 per iteration).

typedef float v4f __attribute__((ext_vector_type(4)));

#define NGRAM_N 3           // NO_REPEAT_NGRAM_SIZE in the reference
#define BAN_LDS_TOKENS 4104 // 16.4 KB of the 320 KB/WGP LDS; holds step+1 tokens
#define COPY_THREADS 256
#define COPY_UNROLL 4       // float4s per thread per tile

// ---------------------------------------------------------------------------
// Kernel 1: streaming copy lprobs -> out.
// Block tile = COPY_THREADS * COPY_UNROLL contiguous float4s (16 KB).
// Within a tile each thread loads 4 independent float4s at stride blockDim
// (every access stays wave-coalesced: 32 lanes x 16B = 512B/clause), then
// stores them. 4 outstanding b128 loads per lane instead of 1.
// ---------------------------------------------------------------------------
__global__ void ngram_copy_kernel(const float* __restrict__ in,
                                  float* __restrict__ out,
                                  long n) {
  const long n4 = n >> 2;
  const v4f* __restrict__ in4 = (const v4f*)in;
  v4f* __restrict__ out4 = (v4f*)out;

  const long tile = (long)COPY_THREADS * COPY_UNROLL;          // float4s per block-tile
  const long grid_stride = (long)gridDim.x * tile;

  for (long base = (long)blockIdx.x * tile; base < n4; base += grid_stride) {
    const long i0 = base + threadIdx.x;

    if (base + tile <= n4) {
      // Full tile: 4 independent NT loads, then 4 NT stores.
      __builtin_prefetch(in4 + i0 + grid_stride, 0, 1);  // gfx1250 global_prefetch_b8
      v4f v0 = __builtin_nontemporal_load(in4 + i0 + 0 * COPY_THREADS);
      v4f v1 = __builtin_nontemporal_load(in4 + i0 + 1 * COPY_THREADS);
      v4f v2 = __builtin_nontemporal_load(in4 + i0 + 2 * COPY_THREADS);
      v4f v3 = __builtin_nontemporal_load(in4 + i0 + 3 * COPY_THREADS);
      __builtin_nontemporal_store(v0, out4 + i0 + 0 * COPY_THREADS);
      __builtin_nontemporal_store(v1, out4 + i0 + 1 * COPY_THREADS);
      __builtin_nontemporal_store(v2, out4 + i0 + 2 * COPY_THREADS);
      __builtin_nontemporal_store(v3, out4 + i0 + 3 * COPY_THREADS);
    } else {
      // Partial last tile.
      #pragma unroll
      for (int k = 0; k < COPY_UNROLL; ++k) {
        const long i = i0 + (long)k * COPY_THREADS;
        if (i < n4) {
          v4f v = __builtin_nontemporal_load(in4 + i);
          __builtin_nontemporal_store(v, out4 + i);
        }
      }
    }
  }

  // Scalar tail (out_size not guaranteed divisible by 4 in general).
  const long base = n4 << 2;
  const long sstride = (long)gridDim.x * (long)blockDim.x;
  long t = base + (long)blockIdx.x * (long)blockDim.x + (long)threadIdx.x;
  for (; t < n; t += sstride) {
    out[t] = __builtin_nontemporal_load(in + t);
  }
}

// ---------------------------------------------------------------------------
// Kernel 2: ban repeated n-grams. One block per row (grid-stride over rows).
// Stage the row's tokens[0..step] in LDS (each token is read by 3 windows),
// then every thread scans windows i, i+blockDim, ... and writes -inf for
// matches. All shape scalars are read on-device (graph-capture safe).
// ---------------------------------------------------------------------------
__global__ void ngram_ban_kernel(const int* __restrict__ tokens,
                                 float* __restrict__ out,
                                 const int* __restrict__ p_bsz,
                                 const int* __restrict__ p_beam,
                                 const int* __restrict__ p_step,
                                 long tokens_flat, long out_flat) {
  __shared__ int sh[BAN_LDS_TOKENS];

  const int bsz  = *p_bsz;
  const int beam = *p_beam;
  const int step = *p_step;

  const int B = bsz * beam;
  if (B <= 0) return;
  const long S = tokens_flat / B;
  const long V = out_flat / B;

  const int num_ngrams = step - NGRAM_N + 2;   // = step - 1 for n=3
  if (num_ngrams <= 0) return;
  const int n_tok = step + 1;                  // tokens[0..step] are referenced
  if (n_tok > (int)S) return;                  // defensive; never true in practice

  const bool use_lds = (n_tok <= BAN_LDS_TOKENS);
  const float neg_inf = __int_as_float(0xFF800000u);

  for (int row = blockIdx.x; row < B; row += gridDim.x) {
    const int* __restrict__ trow = tokens + (long)row * S;
    float* __restrict__ orow = out + (long)row * V;

    if (use_lds) {
      for (int i = threadIdx.x; i < n_tok; i += blockDim.x) sh[i] = trow[i];
      __syncthreads();
    }

    // last (n-1)=2 generated tokens: positions step-1, step
    const int t0 = use_lds ? sh[step - 1] : trow[step - 1];
    const int t1 = use_lds ? sh[step]     : trow[step];

    for (int i = threadIdx.x; i < num_ngrams; i += blockDim.x) {
      const int a = use_lds ? sh[i]     : trow[i];
      const int b = use_lds ? sh[i + 1] : trow[i + 1];
      if (a == t0 && b == t1) {
        const int ban = use_lds ? sh[i + 2] : trow[i + 2];
        if ((unsigned)ban < (unsigned)V) {
          orow[ban] = neg_inf;   // idempotent; no atomic needed
        }
      }
    }
    __syncthreads();  // protect sh[] before the next row refills it
  }
}

// ---------------------------------------------------------------------------
// Launch. Inputs (setup_inputs order): tokens, lprobs, bsz, beam_size, step.
// ---------------------------------------------------------------------------
extern "C" void kernel_launch(void* const* d_in, const int* in_sizes, int n_in,
                              void* d_out, int out_size, void* d_ws, size_t ws_size,
                              hipStream_t stream) {
  const int*   tokens = (const int*)d_in[0];
  const float* lprobs = (const float*)d_in[1];
  const int*   p_bsz  = (const int*)d_in[2];
  const int*   p_beam = (const int*)d_in[3];
  const int*   p_step = (const int*)d_in[4];
  float* out = (float*)d_out;

  const long n = (long)out_size;           // B*V floats (~25.7M)
  const long tokens_flat = (long)in_sizes[0];

  // Copy: one 16KB tile (256 threads x 4 float4s) per block per pass.
  const long n4 = n >> 2;
  const long tile = (long)COPY_THREADS * COPY_UNROLL;
  long cblocks = (n4 + tile - 1) / tile;   // ~6283 blocks for n4=6.43M
  if (cblocks < 1) cblocks = 1;
  if (cblocks > 1048576) cblocks = 1048576;
  ngram_copy_kernel<<<(unsigned)cblocks, COPY_THREADS, 0, stream>>>(lprobs, out, n);

  // Ban: grid-stride over rows; B is only known on-device, so launch a
  // generous fixed grid (idle blocks exit immediately).
  ngram_ban_kernel<<<2048, 256, 0, stream>>>(tokens, out, p_bsz, p_beam, p_step,
                                             tokens_flat, n);
}